// Guido_5634997092362
// MI455X (gfx1250) — compile-verified
//
#include <hip/hip_runtime.h>
#include <hip/hip_bf16.h>
#include <math.h>

// ---------------------------------------------------------------------------
// HKB coupled-oscillator RK4 agent step, MI455X (gfx1250, wave32).
//
// Math rewrite:  s1_i = Sum_j Kp[i,j] sin(p_i - p_j)
//                     = sin(p_i)*(Kp.cos p)_i - cos(p_i)*(Kp.sin p)_i
// and sin(2d) analogously with sin2p = 2 sp cp, cos2p = 1 - 2 sp^2.
// => per RK4 sub-step: 4 sin + 4 cos (TRANS pipe) + four 4x4 mat-vecs.
//
// The mat-vecs run on the matrix pipe with V_WMMA_F32_16X16X4_F32:
//   A (16x4)  = K matrix, wave-uniform, rows 0-3 (variant "lo") or rows 8-11
//               (variant "hi"); built once per kernel from per-lane loads.
//   B (4x16)  = 16 agents' 4-vectors as columns (ds_bpermute marshaling,
//               minimized: identity lanes use the register directly).
//   D (16x16) accumulate: WMMA(A_lo, B[agents 0-15]) then
//               WMMA(A_hi, B[agents 16-31]) onto the same C. D VGPR r then
//               holds component r of (K.v) for agent == lane, all 32 lanes,
//               so there is zero output un-marshaling.
// Pipes: TRANS = sin/cos, MATRIX = contraction, DS = shuffles, VALU = glue.
// ---------------------------------------------------------------------------

typedef float v2f __attribute__((ext_vector_type(2)));
typedef float v8f __attribute__((ext_vector_type(8)));

__device__ __forceinline__ float bperm(float v, int byte_idx) {
    return __int_as_float(__builtin_amdgcn_ds_bpermute(byte_idx, __float_as_int(v)));
}

// Build B (4x16, f32) pair for one per-lane 4-vector v[0..3].
// Assumed B layout (mirror of documented 32-bit A 16x4 layout):
//   VGPR0 = row k=0 (lanes 0-15) | row k=2 (lanes 16-31)
//   VGPR1 = row k=1 (lanes 0-15) | row k=3 (lanes 16-31)
// B1 columns = agents (lanes) 0-15 of this wave, B2 columns = agents 16-31.
// Only 4 bpermutes: lanes whose data is already in place take the register
// directly (B1 low half = own v0/v1; B2 high half = own v2/v3).
__device__ __forceinline__ void build_B(const float v0, const float v1,
                                        const float v2, const float v3,
                                        int lane, v2f& B1, v2f& B2) {
    const bool hi    = (lane & 16) != 0;
    const int  lo_ix = (lane & 15) * 4;          // byte idx: agent (lane&15)
    const int  hi_ix = ((lane & 15) + 16) * 4;   // byte idx: agent (lane&15)+16

    float b1x = bperm(v2, lo_ix);   // used by high lanes: v2 of agent lane-16
    float b1y = bperm(v3, lo_ix);   // used by high lanes: v3 of agent lane-16
    float b2x = bperm(v0, hi_ix);   // used by low lanes:  v0 of agent lane+16
    float b2y = bperm(v1, hi_ix);   // used by low lanes:  v1 of agent lane+16

    B1.x = hi ? b1x : v0;
    B1.y = hi ? b1y : v1;
    B2.x = hi ? v2  : b2x;
    B2.y = hi ? v3  : b2y;
}

// out[i] = (K . v)_i for this lane's agent, via two accumulating WMMAs.
__device__ __forceinline__ void kmatvec(v2f Alo, v2f Ahi,
                                        const float v0, const float v1,
                                        const float v2, const float v3,
                                        int lane, float out[4]) {
    v2f B1, B2;
    build_B(v0, v1, v2, v3, lane, B1, B2);
    v8f acc = {};
    acc = __builtin_amdgcn_wmma_f32_16x16x4_f32(false, Alo, false, B1,
                                                (short)0, acc, false, false);
    acc = __builtin_amdgcn_wmma_f32_16x16x4_f32(false, Ahi, false, B2,
                                                (short)0, acc, false, false);
    out[0] = acc[0]; out[1] = acc[1]; out[2] = acc[2]; out[3] = acc[3];
}

// One HKB evaluation: out = fx - s1 - s2  (fx = freqs + x4, constant per agent)
__device__ __forceinline__ void hkb(const float p[4],
                                    v2f ApLo, v2f ApHi, v2f AaLo, v2f AaHi,
                                    const float fx[4], int lane, float out[4]) {
    float sp[4], cp[4], s2[4], c2[4];
#pragma unroll
    for (int i = 0; i < 4; ++i) {
        sp[i] = __sinf(p[i]);                 // v_sin_f32 (TRANS pipe)
        cp[i] = __cosf(p[i]);                 // v_cos_f32
        s2[i] = 2.0f * sp[i] * cp[i];         // sin 2p
        c2[i] = 1.0f - 2.0f * sp[i] * sp[i];  // cos 2p
    }
    float Mcp[4], Msp[4], Mc2[4], Ms2[4];
    kmatvec(ApLo, ApHi, cp[0], cp[1], cp[2], cp[3], lane, Mcp);
    kmatvec(ApLo, ApHi, sp[0], sp[1], sp[2], sp[3], lane, Msp);
    kmatvec(AaLo, AaHi, c2[0], c2[1], c2[2], c2[3], lane, Mc2);
    kmatvec(AaLo, AaHi, s2[0], s2[1], s2[2], s2[3], lane, Ms2);
#pragma unroll
    for (int i = 0; i < 4; ++i) {
        float s1  = sp[i] * Mcp[i] - cp[i] * Msp[i];
        float s2v = s2[i] * Mc2[i] - c2[i] * Ms2[i];
        out[i] = fx[i] - s1 - s2v;
    }
}

__global__ __launch_bounds__(256)
void guido_rk4_kernel(const float* __restrict__ x_in,     // [B,2]
                      const float* __restrict__ phases,   // [B,4]
                      const float* __restrict__ K_phase,  // [4,4]
                      const float* __restrict__ K_anti,   // [4,4]
                      const float* __restrict__ freqs,    // [4]
                      float* __restrict__ out,            // [B,3]
                      int n) {
    const int tid  = blockIdx.x * blockDim.x + threadIdx.x;
    const int lane = threadIdx.x & 31;
    const bool live = tid < n;
    const int a = live ? tid : (n - 1);   // clamp: keep EXEC full for WMMA

    // ---- wave-uniform A matrices (built from per-lane gather loads) -------
    // A 16x4 f32 layout: lane m & lane 16+m hold row (m&15);
    //   VGPR0 = K-col {0|2}, VGPR1 = K-col {1|3} by lane half.
    const int row  = lane & 15;
    const int colb = ((lane >> 4) & 1) * 2;   // 0 -> cols 0,1 ; 1 -> cols 2,3
    const int r4   = row & 3;
    const float kp0 = K_phase[r4 * 4 + colb + 0];
    const float kp1 = K_phase[r4 * 4 + colb + 1];
    const float ka0 = K_anti [r4 * 4 + colb + 0];
    const float ka1 = K_anti [r4 * 4 + colb + 1];
    const bool in_lo = (row < 4);
    const bool in_hi = (row >= 8) && (row < 12);
    v2f ApLo = { in_lo ? kp0 : 0.0f, in_lo ? kp1 : 0.0f };
    v2f ApHi = { in_hi ? kp0 : 0.0f, in_hi ? kp1 : 0.0f };
    v2f AaLo = { in_lo ? ka0 : 0.0f, in_lo ? ka1 : 0.0f };
    v2f AaHi = { in_hi ? ka0 : 0.0f, in_hi ? ka1 : 0.0f };

    // ---- per-agent streamed loads (b64/b128) ------------------------------
    const float2 x  = ((const float2*)x_in)[a];
    const float4 p4 = ((const float4*)phases)[a];
    const float4 fq = *((const float4*)freqs);

    float p[4]  = { p4.x, p4.y, p4.z, p4.w };
    float fx[4] = { fq.x + x.x, fq.y + x.y, fq.z, fq.w };  // x4 = [x0,x1,0,0]

    // ---- RK4, h = 1/500 ---------------------------------------------------
    const float h = 1.0f / 500.0f;
    float k1[4], k2[4], k3[4], k4[4], t[4];

    hkb(p, ApLo, ApHi, AaLo, AaHi, fx, lane, k1);
#pragma unroll
    for (int i = 0; i < 4; ++i) { k1[i] *= h; t[i] = p[i] + 0.5f * k1[i]; }
    hkb(t, ApLo, ApHi, AaLo, AaHi, fx, lane, k2);
#pragma unroll
    for (int i = 0; i < 4; ++i) { k2[i] *= h; t[i] = p[i] + 0.5f * k2[i]; }
    hkb(t, ApLo, ApHi, AaLo, AaHi, fx, lane, k3);
#pragma unroll
    for (int i = 0; i < 4; ++i) { k3[i] *= h; t[i] = p[i] + k3[i]; }
    hkb(t, ApLo, ApHi, AaLo, AaHi, fx, lane, k4);

    // new_phases components 2,3 (only ones the head consumes)
    const float np2 = (k1[2] + 2.0f * k2[2] + 2.0f * k3[2] + h * k4[2]) * (1.0f / 6.0f);
    const float np3 = (k1[3] + 2.0f * k2[3] + 2.0f * k3[3] + h * k4[3]) * (1.0f / 6.0f);

    // ---- decision head ----------------------------------------------------
    float ang = np3 - np2;
    float m2  = ang - 2.0f * floorf(ang * 0.5f);       // jnp.mod(ang, 2) in [0,2)
    float q   = m2 * ((float)M_PI * 0.25f);            // (ang % 2) * pi / 4
    float sq  = __sinf(q);
    float cq  = __cosf(q);
    float pr  = (sq > 0.0f) ? 1.0f : 0.0f;             // heaviside(sin q, 0)
    float pl  = (sq < 0.0f) ? 1.0f : 0.0f;             // heaviside(-sin q, 0)
    float pf  = cq;
    float mx  = fmaxf(pr, fmaxf(pl, pf));
    float e0 = __expf(pr - mx), e1 = __expf(pl - mx), e2 = __expf(pf - mx);
    float inv = 1.0f / (e0 + e1 + e2);

    if (live) {
        out[3 * a + 0] = e0 * inv;
        out[3 * a + 1] = e1 * inv;
        out[3 * a + 2] = e2 * inv;
    }
}

extern "C" void kernel_launch(void* const* d_in, const int* in_sizes, int n_in,
                              void* d_out, int out_size, void* d_ws, size_t ws_size,
                              hipStream_t stream) {
    (void)n_in; (void)d_ws; (void)ws_size; (void)out_size;
    const float* x_in    = (const float*)d_in[0];
    const float* phases  = (const float*)d_in[1];
    const float* K_phase = (const float*)d_in[2];
    const float* K_anti  = (const float*)d_in[3];
    const float* freqs   = (const float*)d_in[4];
    float* out = (float*)d_out;

    const int n = in_sizes[1] / 4;                 // B agents (phases is [B,4])
    const int threads = 256;                       // 8 wave32 per block
    const int blocks  = (n + threads - 1) / threads;
    guido_rk4_kernel<<<blocks, threads, 0, stream>>>(x_in, phases, K_phase,
                                                     K_anti, freqs, out, n);
}